// CustomANFIS_4922032521302
// MI455X (gfx1250) — compile-verified
//
#include <hip/hip_runtime.h>

// ANFIS fused forward for MI455X (gfx1250, wave32).
// N=4096 samples, D=128 dims, R=256 rules, O=64 outputs.
// out[n,j] = softmax_j( sum_{r,i} (norm[n,r]*X[n,i])*coeffs[r,i,j] + sum_r norm[n,r]*bias[r,j] )

#define N_ 4096
#define D_ 128
#define R_ 256
#define O_ 64
#define CR_ ((D_ + 1) * O_)   // 8256 floats per rule in coeffs
#define TILE_M 16             // samples per workgroup
#define NP (R_ + 2)           // padded norm row stride (bank-conflict free)

typedef float v2f __attribute__((ext_vector_type(2)));
typedef float v8f __attribute__((ext_vector_type(8)));

__global__ __launch_bounds__(256)
void anfis_fused_kernel(const float* __restrict__ X,
                        const float* __restrict__ centers,
                        const float* __restrict__ sigmas,
                        const float* __restrict__ coeffs,
                        float* __restrict__ out)
{
    __shared__ float x_lds[D_][TILE_M];       // X tile transposed [d][n]   (8 KB)
    __shared__ float nm_lds[TILE_M][NP];      // strengths -> norm          (~16.5 KB)
    __shared__ float o2_lds[2][TILE_M][O_];   // partial output tiles       (8 KB)
    __shared__ float rinv[TILE_M];

    const int t     = threadIdx.x;   // 0..255  (8 waves)
    const int wave  = t >> 5;        // 0..7
    const int lane  = t & 31;
    const int grp   = wave >> 2;     // rule-split group: 0 or 1
    const int j0    = (wave & 3) * 16;      // output column tile
    const int mrow  = lane & 15;            // A row M / B col N for this lane
    const int khalf = (lane >> 4) * 2;      // lanes 0-15: K+0,K+1 ; 16-31: K+2,K+3
    const int n0    = blockIdx.x * TILE_M;

    // ---------------- load X tile (transposed for broadcast reads) ----------------
    for (int i = t; i < TILE_M * D_; i += 256) {
        const int n = i / D_, d = i % D_;
        x_lds[d][n] = X[(size_t)(n0 + n) * D_ + d];
    }
    __syncthreads();

    // ---------------- layer 1+2: log-strengths, one rule per thread ----------------
    {
        const int r = t;
        float acc[TILE_M];
        #pragma unroll
        for (int n = 0; n < TILE_M; ++n) acc[n] = 0.0f;
        for (int d = 0; d < D_; ++d) {
            const float c = centers[(size_t)r * D_ + d];
            const float s = sigmas[(size_t)r * D_ + d];
            const float w = 1.0f / (2.0f * s * s);
            #pragma unroll
            for (int n = 0; n < TILE_M; ++n) {
                const float dx = x_lds[d][n] - c;
                acc[n] -= dx * dx * w;
            }
        }
        #pragma unroll
        for (int n = 0; n < TILE_M; ++n)
            nm_lds[n][r] = __expf(acc[n]);       // strength
    }
    __syncthreads();

    // ---------------- layer 3: normalize across rules ----------------
    if (t < TILE_M) {
        float s = 0.0f;
        for (int r = 0; r < R_; ++r) s += nm_lds[t][r];
        rinv[t] = 1.0f / (s + 1e-8f);
    }
    __syncthreads();
    {
        const int r = t;
        #pragma unroll
        for (int n = 0; n < TILE_M; ++n)
            nm_lds[n][r] *= rinv[n];
    }

    // hoist this lane's X fragments into VGPRs (fully static indexing -> no scratch)
    v2f xf[D_ / 4];                          // 32 x v2f = 64 VGPRs
    #pragma unroll
    for (int i = 0; i < D_ / 4; ++i) {
        v2f v;
        v.x = x_lds[4 * i + khalf][mrow];
        v.y = x_lds[4 * i + khalf + 1][mrow];
        xf[i] = v;
    }
    __syncthreads();

    // ---------------- layers 4+5: fp32 WMMA GEMM, no barriers in the loop ----------------
    v8f acc = {};                            // 16x16 f32 C/D tile (8 VGPRs)

    for (int r = grp; r < R_; r += 2) {      // wave-groups split the rules
        const float  nrm = nm_lds[mrow][r];  // conflict-free broadcast (padded stride)
        const float* Cr  = coeffs + (size_t)r * CR_ + j0 + mrow;
        #pragma unroll                        // FULL unroll: xf stays in registers
        for (int i = 0; i < D_ / 4; ++i) {
            v2f a, b;
            a.x = nrm * xf[i].x;             // -> v_pk_mul_f32, co-executes with XDL
            a.y = nrm * xf[i].y;
            b.x = Cr[(size_t)(4 * i + khalf + 0) * O_];   // L2-resident coeffs
            b.y = Cr[(size_t)(4 * i + khalf + 1) * O_];
            acc = __builtin_amdgcn_wmma_f32_16x16x4_f32(
                false, a, false, b, (short)0, acc, false, false);
        }
    }

    // bias GEMM: acc += norm[16x256] @ bias[256x64], K-range split across groups
    {
        const v2f*   nrow = (const v2f*)&nm_lds[mrow][0];
        const float* Bias = coeffs + (size_t)D_ * O_ + j0 + mrow;   // coeffs[r, D, j]
        const int    kb   = grp * (R_ / 2);
        #pragma unroll 8
        for (int k0 = 0; k0 < R_ / 2; k0 += 4) {
            const int k = kb + k0 + khalf;   // even -> 8B-aligned b64 LDS load
            v2f a = nrow[k >> 1];
            v2f b;
            b.x = Bias[(size_t)(k + 0) * CR_];
            b.y = Bias[(size_t)(k + 1) * CR_];
            acc = __builtin_amdgcn_wmma_f32_16x16x4_f32(
                false, a, false, b, (short)0, acc, false, false);
        }
    }

    // spill C/D tile: lane holds column N=lane&15, VGPR v holds row M = v + 8*(lane>=16)
    #pragma unroll
    for (int v = 0; v < 8; ++v) {
        const int m = v + 8 * (lane >> 4);
        o2_lds[grp][m][j0 + mrow] = acc[v];
    }
    __syncthreads();

    // ---------------- reduce the two rule-groups + softmax over 64 outputs ----------------
    if (t < TILE_M) {
        float mx = -1e30f;
        for (int j = 0; j < O_; ++j)
            mx = fmaxf(mx, o2_lds[0][t][j] + o2_lds[1][t][j]);
        float s = 0.0f;
        for (int j = 0; j < O_; ++j)
            s += __expf(o2_lds[0][t][j] + o2_lds[1][t][j] - mx);
        const float inv = 1.0f / s;
        for (int j = 0; j < O_; ++j)
            out[(size_t)(n0 + t) * O_ + j] =
                __expf(o2_lds[0][t][j] + o2_lds[1][t][j] - mx) * inv;
    }
}

extern "C" void kernel_launch(void* const* d_in, const int* in_sizes, int n_in,
                              void* d_out, int out_size, void* d_ws, size_t ws_size,
                              hipStream_t stream) {
    (void)in_sizes; (void)n_in; (void)d_ws; (void)ws_size; (void)out_size;
    const float* X       = (const float*)d_in[0];
    const float* centers = (const float*)d_in[1];
    const float* sigmas  = (const float*)d_in[2];
    const float* coeffs  = (const float*)d_in[3];
    float* out = (float*)d_out;

    dim3 grid(N_ / TILE_M);   // 256 workgroups
    dim3 block(256);          // 8 waves (wave32), 2 waves/SIMD
    anfis_fused_kernel<<<grid, block, 0, stream>>>(X, centers, sigmas, coeffs, out);
}